// RelativePositionalEncoding_25666724560954
// MI455X (gfx1250) — compile-verified
//
#include <hip/hip_runtime.h>
#include <hip/hip_bf16.h>

typedef __attribute__((ext_vector_type(2))) float v2f;
typedef __attribute__((ext_vector_type(8))) float v8f;

#define B_   8
#define S_   2048
#define D_   1024
#define M_   128
#define R_   257      // 2*M+1
#define RP_  260      // R padded to multiple of 4 (K dim for WMMA)
#define INV_S (1.0f / (float)S_)

// ---------------------------------------------------------------------------
// Kernel 1: build the (batch-independent) relative-position counts matrix,
// zero-padded in K to RP_.  counts[i][r]:
//   r == 0    : #{a : i - a <= -M}  = max(0, S - i - M)
//   r == 2M   : #{a : i - a >=  M}  = max(0, i - M + 1)
//   1..2M-1   : 1 iff a = i + M - r in [0, S)
//   r >= R    : 0 (padding)
// ---------------------------------------------------------------------------
__global__ __launch_bounds__(256) void build_counts(float* __restrict__ counts) {
    int idx = blockIdx.x * 256 + threadIdx.x;
    if (idx >= S_ * RP_) return;
    int i = idx / RP_;
    int r = idx % RP_;
    float v;
    if (r == 0) {
        int c = S_ - i - M_;
        v = (float)(c > 0 ? c : 0);
    } else if (r == 2 * M_) {
        int c = i - M_ + 1;
        v = (float)(c > 0 ? c : 0);
    } else if (r < 2 * M_) {
        int a = i + M_ - r;
        v = (a >= 0 && a < S_) ? 1.0f : 0.0f;
    } else {
        v = 0.0f;   // K padding
    }
    counts[idx] = v;
}

// ---------------------------------------------------------------------------
// Kernel 1b: transpose + zero-pad table into tT[d][r] (D x RP_) so the WMMA
// B fragment (two consecutive K values per lane) is one aligned b64 load with
// no range guards (pad rows read as 0, matching the zero counts columns).
// ---------------------------------------------------------------------------
__global__ __launch_bounds__(256) void build_tT(const float* __restrict__ table,
                                                float* __restrict__ tT) {
    int idx = blockIdx.x * 256 + threadIdx.x;
    if (idx >= D_ * RP_) return;
    int d = idx / RP_;
    int r = idx % RP_;
    tT[idx] = (r < R_) ? table[r * D_ + d] : 0.0f;
}

// ---------------------------------------------------------------------------
// Kernel 1c: fold 1/S and the (S-1) residual coefficient into the conv taps:
//   wt[0*D+d] = w[d][0]/S,  wt[1*D+d] = (w[d][1]+S-1)/S,  wt[2*D+d] = w[d][2]/S
// ---------------------------------------------------------------------------
__global__ __launch_bounds__(256) void build_wt(const float* __restrict__ w,
                                                float* __restrict__ wt) {
    int d = blockIdx.x * 256 + threadIdx.x;   // D threads
    if (d >= D_) return;
    wt[0 * D_ + d] = w[d * 3 + 0] * INV_S;
    wt[1 * D_ + d] = (w[d * 3 + 1] + (float)(S_ - 1)) * INV_S;
    wt[2 * D_ + d] = w[d * 3 + 2] * INV_S;
}

// ---------------------------------------------------------------------------
// Kernel 2: emb(S x D) = counts(S x RP_) @ table(RP_ x D) via
// V_WMMA_F32_16X16X4_F32.  One 16x16 C tile per wave32, 8 waves per block.
// A 16x4 layout : lanes 0-15 -> K = k,k+1 in v0,v1 ; lanes 16-31 -> K = k+2,k+3
// B 4x16 layout : lane%16 = column, same half/K split as A (read from tT)
// C layout      : VGPR v, M = v + 8*(lane/16), N = lane%16
// Inner loop: two unconditional global_load_b64 + one wmma; EXEC all-ones.
// ---------------------------------------------------------------------------
__global__ __launch_bounds__(256) void wmma_emb(const float* __restrict__ counts,
                                                const float* __restrict__ tT,
                                                float* __restrict__ emb) {
    const int lane = threadIdx.x & 31;
    const int wave = threadIdx.x >> 5;
    const int tile = blockIdx.x * 8 + wave;      // (S/16)*(D/16) = 8192 tiles
    const int tiles_col = D_ / 16;               // 64
    const int row0 = (tile / tiles_col) * 16;
    const int col0 = (tile % tiles_col) * 16;
    const int half = lane >> 4;                  // 0 or 1
    const int lm   = lane & 15;

    const float* arow = counts + (size_t)(row0 + lm) * RP_ + 2 * half;
    const float* bcol = tT     + (size_t)(col0 + lm) * RP_ + 2 * half;

    v8f c = {};
    for (int k = 0; k < RP_; k += 4) {
        v2f a = *(const v2f*)(arow + k);   // counts[row0+lm][k+2h .. +1]
        v2f b = *(const v2f*)(bcol + k);   // table [k+2h .. +1][col0+lm]
        c = __builtin_amdgcn_wmma_f32_16x16x4_f32(false, a, false, b,
                                                  (short)0, c, false, false);
    }
#pragma unroll
    for (int v = 0; v < 8; ++v) {
        emb[(size_t)(row0 + v + 8 * half) * D_ + col0 + lm] = c[v];
    }
}

// ---------------------------------------------------------------------------
// Kernel 3: fold all batch-independent terms into pre[i][d] (in place on emb):
// pre = (bias - table[M] + w1*table[M] + [i>=1]*w0*table[M-1]
//        + [i<=S-2]*w2*table[M+1] + emb) / S
// ---------------------------------------------------------------------------
__global__ __launch_bounds__(256) void make_pre(float* __restrict__ pre,
                                                const float* __restrict__ table,
                                                const float* __restrict__ w,
                                                const float* __restrict__ bias) {
    const int i = blockIdx.x;                          // S
    const int d = blockIdx.y * 256 + threadIdx.x;      // D (grid.y = 4)
    const float e   = pre[(size_t)i * D_ + d];
    const float tm1 = table[(M_ - 1) * D_ + d];
    const float t0  = table[(M_    ) * D_ + d];
    const float tp1 = table[(M_ + 1) * D_ + d];
    const float w0  = w[d * 3 + 0];
    const float w1  = w[d * 3 + 1];
    const float w2  = w[d * 3 + 2];
    float acc = bias[d] - t0 + w1 * t0 + e;
    if (i >= 1)      acc += w0 * tm1;
    if (i <= S_ - 2) acc += w2 * tp1;
    pre[(size_t)i * D_ + d] = acc * INV_S;
}

// ---------------------------------------------------------------------------
// Kernel 4: bandwidth-bound streaming kernel (the 128 MB of real traffic).
// out[b,i,d] = pre[i,d] + wt0[d]*x[b,i-1,d] + wt1[d]*x[b,i,d] + wt2[d]*x[b,i+1,d]
// float4 over D; neighbor rows of x and all of pre/wt hit the 192 MB L2.
// ---------------------------------------------------------------------------
__global__ __launch_bounds__(256) void main_conv(const float* __restrict__ x,
                                                 const float* __restrict__ pre,
                                                 const float* __restrict__ wt,
                                                 float* __restrict__ out) {
    const int i = blockIdx.x;     // S
    const int b = blockIdx.y;     // B
    const int t = threadIdx.x;    // 256 float4 lanes over D

    const float4* x4  = (const float4*)x;
    const float4* p4  = (const float4*)pre;
    const float4* wt4 = (const float4*)wt;
    float4*       o4  = (float4*)out;

    const long rowc = ((long)b * S_ + i) * (D_ / 4) + t;
    const float4 xc = x4[rowc];
    float4 xl = make_float4(0.f, 0.f, 0.f, 0.f);
    float4 xr = make_float4(0.f, 0.f, 0.f, 0.f);
    if (i > 0)      xl = x4[rowc - (D_ / 4)];
    if (i < S_ - 1) xr = x4[rowc + (D_ / 4)];
    const float4 p  = p4[(long)i * (D_ / 4) + t];
    const float4 c0 = wt4[0 * (D_ / 4) + t];
    const float4 c1 = wt4[1 * (D_ / 4) + t];
    const float4 c2 = wt4[2 * (D_ / 4) + t];

    float4 o;
    o.x = p.x + c0.x * xl.x + c1.x * xc.x + c2.x * xr.x;
    o.y = p.y + c0.y * xl.y + c1.y * xc.y + c2.y * xr.y;
    o.z = p.z + c0.z * xl.z + c1.z * xc.z + c2.z * xr.z;
    o.w = p.w + c0.w * xl.w + c1.w * xc.w + c2.w * xr.w;
    o4[rowc] = o;
}

extern "C" void kernel_launch(void* const* d_in, const int* in_sizes, int n_in,
                              void* d_out, int out_size, void* d_ws, size_t ws_size,
                              hipStream_t stream) {
    const float* x     = (const float*)d_in[0];   // (B, S, D)
    const float* table = (const float*)d_in[1];   // (R, D)
    const float* w     = (const float*)d_in[2];   // (D, 3)
    const float* bias  = (const float*)d_in[3];   // (D,)
    float*       out   = (float*)d_out;           // (B, S, D)

    // Workspace layout (floats):
    //   [counts: S*RP_][tT: D*RP_][wt: 3*D][emb/pre: S*D]  ~ 11.6 MB total
    float* counts = (float*)d_ws;
    float* tT     = counts + (size_t)S_ * RP_;
    float* wt     = tT     + (size_t)D_ * RP_;
    float* emb    = wt     + (size_t)3 * D_;

    build_counts<<<(S_ * RP_ + 255) / 256, 256, 0, stream>>>(counts);
    build_tT   <<<(D_ * RP_ + 255) / 256, 256, 0, stream>>>(table, tT);
    build_wt   <<<(D_ + 255) / 256, 256, 0, stream>>>(w, wt);

    wmma_emb<<<(S_ / 16) * (D_ / 16) / 8, 256, 0, stream>>>(counts, tT, emb);

    dim3 gpre(S_, D_ / 256);
    make_pre<<<gpre, 256, 0, stream>>>(emb, table, w, bias);

    dim3 gmain(S_, B_);
    main_conv<<<gmain, 256, 0, stream>>>(x, emb, wt, out);
}